// VectorQuantizer_53008486367572
// MI455X (gfx1250) — compile-verified
//
#include <hip/hip_runtime.h>

typedef __attribute__((ext_vector_type(2))) float v2f;
typedef __attribute__((ext_vector_type(8))) float v8f;

#define NUM_EMB 512
#define EDIM    64
#define DIM_B   4
#define DIM_D   16
#define DIM_H   64
#define DIM_W   64
#define N_TOTAL (DIM_B*DIM_D*DIM_H*DIM_W)   // 262144 spatial vectors
#define CSTRIDE (DIM_D*DIM_H*DIM_W)         // 65536 elements between channels
#define BSTRIDE (EDIM*CSTRIDE)              // 4194304 elements between batches
#define ROWS_PER_BLOCK 128
#define LDSPAD  66                          // 64 + 2 pad (keeps 8B align, dodges bank conflicts)

// ---------------- Kernel 1: half codebook norms + zero usage mask ----------------
__global__ __launch_bounds__(512) void vq_prep(const float* __restrict__ emb,
                                               float* __restrict__ halfnorm,
                                               unsigned* __restrict__ gmask) {
    const int k = threadIdx.x;              // 512 threads, one per code
    if (k < 16) gmask[k] = 0u;
    float s = 0.f;
    #pragma unroll
    for (int c = 0; c < EDIM; ++c) {
        float e = emb[k * EDIM + c];
        s += e * e;
    }
    halfnorm[k] = 0.5f * s;
}

// ---------------- Kernel 2: WMMA distance GEMM + argmin + gather ----------------
__global__ __launch_bounds__(256) void vq_main(const float* __restrict__ z,
                                               const float* __restrict__ emb,
                                               const float* __restrict__ halfnorm,
                                               unsigned* __restrict__ gmask,
                                               float* __restrict__ out) {
    __shared__ float lds_z[ROWS_PER_BLOCK][LDSPAD];   // 33792 B
    __shared__ float lds_e[64][LDSPAD];               // 16896 B
    __shared__ int   idx_s[ROWS_PER_BLOCK];
    __shared__ unsigned bmask[16];

    const int tid  = threadIdx.x;
    const int lane = tid & 31;
    const int wave = tid >> 5;
    const int n0   = blockIdx.x * ROWS_PER_BLOCK;     // first spatial row of this block
    const int b    = n0 >> 16;                        // 65536 rows per batch
    const int d    = (n0 >> 12) & 15;                 // 4096 rows per depth slice
    const int hw0  = n0 & 4095;
    const long gbase = (long)b * BSTRIDE + (long)d * (DIM_H * DIM_W) + hw0;

    if (tid < 16) bmask[tid] = 0u;

    // Stage Z tile: lds_z[row][chan] = z[gbase + chan*CSTRIDE + row] (coalesced per chan)
    #pragma unroll
    for (int i = tid; i < ROWS_PER_BLOCK * EDIM; i += 256) {
        int c = i >> 7;          // / 128
        int r = i & 127;
        lds_z[r][c] = z[gbase + (long)c * CSTRIDE + r];
    }
    __syncthreads();

    // Preload all 16 A-fragments (this wave's 16 rows, full K=64).
    // V_WMMA_F32_16X16X4_F32 A layout: lanes 0-15 hold M=0..15 K={0,1}; lanes 16-31 K={2,3}.
    const int m       = lane & 15;
    const int koff    = (lane >> 4) << 1;   // 0 or 2
    const int rowbase = wave << 4;
    v2f afrag[16];
    #pragma unroll
    for (int k = 0; k < 16; ++k) {
        afrag[k][0] = lds_z[rowbase + m][4 * k + koff];
        afrag[k][1] = lds_z[rowbase + m][4 * k + koff + 1];
    }

    float bestv[8];
    int   besti[8];
    #pragma unroll
    for (int i = 0; i < 8; ++i) { bestv[i] = 3.4e38f; besti[i] = 0; }

    for (int tile = 0; tile < 8; ++tile) {           // 8 tiles of 64 codes
        __syncthreads();                              // protect lds_e reuse
        #pragma unroll
        for (int i = tid; i < 64 * EDIM; i += 256)
            lds_e[i >> 6][i & 63] = emb[tile * 4096 + i];   // coalesced
        if (tile < 7)
            __builtin_prefetch(&emb[(tile + 1) * 4096], 0, 1);  // global_prefetch_b8
        __syncthreads();

        #pragma unroll
        for (int cs = 0; cs < 4; ++cs) {             // 4 column sub-tiles of 16 codes
            const int col_local = (cs << 4) + m;     // B = E^T: same lane pattern as A
            v8f acc = {};
            #pragma unroll
            for (int k = 0; k < 16; ++k) {           // K = 64 in steps of 4
                v2f bfrag;
                bfrag[0] = lds_e[col_local][4 * k + koff];
                bfrag[1] = lds_e[col_local][4 * k + koff + 1];
                acc = __builtin_amdgcn_wmma_f32_16x16x4_f32(
                    false, afrag[k], false, bfrag, (short)0, acc, false, false);
            }
            // D layout: VGPR i -> row rowbase+i (lanes 0-15) / rowbase+i+8 (lanes 16-31),
            // column = lane&15 -> this lane's code. argmin of halfnorm - score.
            const int   mycode = (tile << 6) + (cs << 4) + m;
            const float hn     = halfnorm[mycode];
            #pragma unroll
            for (int i = 0; i < 8; ++i) {
                float dv = hn - acc[i];
                if (dv < bestv[i]) { bestv[i] = dv; besti[i] = mycode; }
            }
        }
    }

    // Cross-lane argmin within each 16-lane half (xor masks stay inside the half).
    #pragma unroll
    for (int off = 8; off > 0; off >>= 1) {
        #pragma unroll
        for (int i = 0; i < 8; ++i) {
            float ov = __shfl_xor(bestv[i], off, 32);
            int   oi = __shfl_xor(besti[i], off, 32);
            if (ov < bestv[i] || (ov == bestv[i] && oi < besti[i])) {
                bestv[i] = ov; besti[i] = oi;
            }
        }
    }
    if ((lane & 15) == 0) {
        const int mb = rowbase + ((lane >> 4) << 3);   // lane 0 -> rows 0..7, lane 16 -> 8..15
        #pragma unroll
        for (int i = 0; i < 8; ++i) {
            idx_s[mb + i] = besti[i];
            atomicOr(&bmask[besti[i] >> 5], 1u << (besti[i] & 31));  // ds_or
        }
    }
    __syncthreads();

    if (tid < 16) atomicOr(&gmask[tid], bmask[tid]);   // one global atomic per word per block

    // Gather + transposed store: out[gbase + c*CSTRIDE + r] = emb[idx[r]][c] (coalesced per chan)
    #pragma unroll
    for (int i = tid; i < ROWS_PER_BLOCK * EDIM; i += 256) {
        int c = i >> 7;
        int r = i & 127;
        out[gbase + (long)c * CSTRIDE + r] = emb[idx_s[r] * EDIM + c];
    }
}

// ---------------- Kernel 3: unique-code count ----------------
__global__ void vq_count(const unsigned* __restrict__ gmask, float* __restrict__ out_unique) {
    if (threadIdx.x == 0) {
        int s = 0;
        #pragma unroll
        for (int w = 0; w < 16; ++w) s += __popc(gmask[w]);
        out_unique[0] = (float)s;
    }
}

extern "C" void kernel_launch(void* const* d_in, const int* in_sizes, int n_in,
                              void* d_out, int out_size, void* d_ws, size_t ws_size,
                              hipStream_t stream) {
    const float* z   = (const float*)d_in[0];   // [4,64,16,64,64] f32
    const float* emb = (const float*)d_in[1];   // [512,64] f32
    float* out = (float*)d_out;                 // 16777216 z_q floats + 1 unique count

    float*    halfnorm = (float*)d_ws;                               // 512 floats
    unsigned* gmask    = (unsigned*)((char*)d_ws + NUM_EMB * sizeof(float)); // 16 words

    vq_prep<<<1, 512, 0, stream>>>(emb, halfnorm, gmask);
    vq_main<<<N_TOTAL / ROWS_PER_BLOCK, 256, 0, stream>>>(z, emb, halfnorm, gmask, out);
    vq_count<<<1, 32, 0, stream>>>(gmask, out + (long)N_TOTAL * EDIM);
}